// LogisticDiscriminantLoss_4157528343145
// MI455X (gfx1250) — compile-verified
//
#include <hip/hip_runtime.h>

typedef __attribute__((ext_vector_type(2))) float v2f;
typedef __attribute__((ext_vector_type(8))) float v8f;

#define EMB_D 256
#define WAVES_PER_BLOCK 8

__global__ void ldml_init_acc(double* acc) {
    if (threadIdx.x < 2) acc[threadIdx.x] = 0.0;
}

__device__ __forceinline__ float stable_softplus(float z) {
    // softplus(z) = max(z,0) + log1p(exp(-|z|))
    return fmaxf(z, 0.0f) + log1pf(__expf(-fabsf(z)));
}

// One wave32 handles 16 pairs. For the 16-pair tile let A = diff matrix (16 x D).
// sqdist = diag(A * A^T), accumulated K=4 at a time with V_WMMA_F32_16X16X4_F32.
// The f32 16x4 A layout and 4x16 B layout are lane-symmetric, so the same
// register pair is passed as both A and B (== A^T).
__global__ __launch_bounds__(256) void ldml_pair_loss_wmma(
    const float* __restrict__ X,
    const float* __restrict__ bias,
    const int*   __restrict__ idx,
    int P, float sign,
    double* __restrict__ acc)
{
    const int lane  = threadIdx.x & 31;
    const int wib   = threadIdx.x >> 5;
    const int wave  = blockIdx.x * WAVES_PER_BLOCK + wib;
    const int pairBase = wave * 16;

    // pair owned by this lane (lanes 16..31 mirror lanes 0..15)
    int p  = pairBase + (lane & 15);
    int pc = (p < P) ? p : 0;                 // clamp, no branch -> EXEC stays all-1s
    const int i = idx[2 * pc];
    const int j = idx[2 * pc + 1];
    const float* __restrict__ rowI = X + (size_t)i * EMB_D;
    const float* __restrict__ rowJ = X + (size_t)j * EMB_D;

    // A-matrix 16x4 f32 layout: lanes 0-15 hold K={k,k+1}, lanes 16-31 hold K={k+2,k+3}
    const int c0 = (lane >> 4) << 1;          // 0 or 2

    v8f c = {};                               // 16x16 f32 accumulator tile
    #pragma unroll 4
    for (int k0 = 0; k0 < EMB_D; k0 += 4) {
        v2f a = *(const v2f*)(rowI + k0 + c0);
        v2f b = *(const v2f*)(rowJ + k0 + c0);
        v2f d = a - b;
        // 8 args: (neg_a, A, neg_b, B, c_mod, C, reuse_a, reuse_b)
        c = __builtin_amdgcn_wmma_f32_16x16x4_f32(
                false, d, false, d, (short)0, c, false, false);
    }

    // Diagonal of the 16x16 C tile:
    //   diag p (0..7)  -> VGPR p,   lane p
    //   diag p (8..15) -> VGPR p-8, lane p+16  (i.e. lanes 24..31, component lane-24)
    union { v8f v; float f[8]; } u; u.v = c;
    float contrib = 0.0f;
    const float b0 = bias[0];
    if (lane < 8 || lane >= 24) {
        const int pd = pairBase + ((lane < 8) ? lane : (lane - 16));
        if (pd < P) {
            const float dist = u.f[(lane < 8) ? lane : (lane - 24)];
            contrib = stable_softplus(sign * (b0 - dist));
        }
    }

    // wave32 butterfly reduce
    #pragma unroll
    for (int off = 16; off >= 1; off >>= 1)
        contrib += __shfl_xor(contrib, off, 32);

    __shared__ float wsum[WAVES_PER_BLOCK];
    if (lane == 0) wsum[wib] = contrib;
    __syncthreads();
    if (threadIdx.x == 0) {
        double s = 0.0;
        #pragma unroll
        for (int w = 0; w < WAVES_PER_BLOCK; ++w) s += (double)wsum[w];
        atomicAdd(acc, s);                    // global_atomic_add_f64
    }
}

__global__ void ldml_finalize(const double* __restrict__ acc,
                              float* __restrict__ out, double invP) {
    if (threadIdx.x < 2) out[threadIdx.x] = (float)(acc[threadIdx.x] * invP);
}

extern "C" void kernel_launch(void* const* d_in, const int* in_sizes, int n_in,
                              void* d_out, int out_size, void* d_ws, size_t ws_size,
                              hipStream_t stream) {
    const float* X    = (const float*)d_in[0];
    const float* bias = (const float*)d_in[1];
    const int*   pos  = (const int*)d_in[2];
    const int*   neg  = (const int*)d_in[3];
    const int P = in_sizes[2] / 2;            // 258048

    double* acc = (double*)d_ws;              // [0]=pos sum, [1]=neg sum
    float*  out = (float*)d_out;

    ldml_init_acc<<<1, 32, 0, stream>>>(acc);

    const int waves  = (P + 15) / 16;         // 16 pairs per wave
    const int blocks = (waves + WAVES_PER_BLOCK - 1) / WAVES_PER_BLOCK;

    // pos: softplus(-(b-d)) -> sign = -1 ; neg: softplus(b-d) -> sign = +1
    ldml_pair_loss_wmma<<<blocks, 256, 0, stream>>>(X, bias, pos, P, -1.0f, acc + 0);
    ldml_pair_loss_wmma<<<blocks, 256, 0, stream>>>(X, bias, neg, P, +1.0f, acc + 1);

    ldml_finalize<<<1, 32, 0, stream>>>(acc, out, 1.0 / (double)P);
}